// Closs_explained_16028817948842
// MI455X (gfx1250) — compile-verified
//
#include <hip/hip_runtime.h>
#include <hip/hip_bf16.h>
#include <stdint.h>

#define N_STOCKS 4096
#define HALF_N   2048
#define BLOCK    256
#define PER_THR  16   // N_STOCKS / BLOCK

typedef __attribute__((ext_vector_type(2))) float        v2f;
typedef __attribute__((ext_vector_type(8))) float        v8f;
typedef __attribute__((ext_vector_type(4))) unsigned int u32x4;
typedef __attribute__((ext_vector_type(8))) int          i32x8;
typedef __attribute__((ext_vector_type(4))) int          i32x4;

#if defined(__gfx1250__) && __has_builtin(__builtin_amdgcn_wmma_f32_16x16x4_f32)
#define HAVE_WMMA_F32 1
#endif
#if defined(__gfx1250__) && __has_builtin(__builtin_amdgcn_tensor_load_to_lds) && \
    __has_builtin(__builtin_amdgcn_s_wait_tensorcnt)
#define HAVE_TDM 1
#endif

// ---------------------------------------------------------------------------
// Block reduction of 256 f32 partials using V_WMMA_F32_16X16X4_F32.
// View the 256 values as a 16x16 matrix M (value of thread t at M[t/16][t%16]).
// With B = all-ones (layout independent), 4 chained K=4 WMMAs give
// D[r][c] = sum_k M[r][k] (row sums, replicated across columns).
// D layout (documented): VGPR v, lanes 0-15 -> row v, lanes 16-31 -> row v+8.
// Result valid on all lanes of wave 0.
// ---------------------------------------------------------------------------
__device__ __forceinline__ float block_reduce_256(float v, float* part) {
  const int t = threadIdx.x;
  part[t] = v;
  __syncthreads();
  float total = 0.0f;
#if defined(HAVE_WMMA_F32)
  if (t < 32) {                       // wave 0, EXEC all-ones as WMMA requires
    const int lane  = t;
    const int r     = lane & 15;      // A rows: lanes 0-15 and 16-31 both map M=0..15
    const int kbase = (lane < 16) ? 0 : 2;  // A 16x4 layout: v0/v1 = K0/K1 (lo lanes), K2/K3 (hi lanes)
    v8f acc = {0.f, 0.f, 0.f, 0.f, 0.f, 0.f, 0.f, 0.f};
    v2f ones;
    ones[0] = 1.0f;
    ones[1] = 1.0f;
#pragma unroll
    for (int s = 0; s < 4; ++s) {     // K-slabs 0..3 of the 16-wide sum
      v2f a;
      a[0] = part[r * 16 + 4 * s + kbase];
      a[1] = part[r * 16 + 4 * s + kbase + 1];
      acc = __builtin_amdgcn_wmma_f32_16x16x4_f32(false, a, false, ones,
                                                  (short)0, acc, false, false);
    }
    // lane<16 holds rows 0..7 in acc[0..7]; lane>=16 holds rows 8..15
    float s8 = acc[0] + acc[1] + acc[2] + acc[3] + acc[4] + acc[5] + acc[6] + acc[7];
    total = s8 + __shfl_xor(s8, 16, 32);
  }
#else
  if (t < 32) {
    float s = 0.f;
#pragma unroll
    for (int k = 0; k < 8; ++k) s += part[t + 32 * k];
#pragma unroll
    for (int d = 16; d; d >>= 1) s += __shfl_xor(s, d, 32);
    total = s;
  }
#endif
  return total;
}

// wave32 inclusive scan via shfl_up
__device__ __forceinline__ float wave_incl_scan(float x) {
  const int lane = threadIdx.x & 31;
#pragma unroll
  for (int d = 1; d < 32; d <<= 1) {
    float u = __shfl_up(x, d, 32);
    if (lane >= d) x += u;
  }
  return x;
}

// ---------------------------------------------------------------------------
// One workgroup (256 threads = 8 wave32) per row of f[8192][4096].
// Row is DMA'd into LDS by the Tensor Data Mover, then a hierarchical
// prefix scan of exp(f)/exp(-f) is built in LDS and the 2048 window/log
// terms are evaluated. Per-row result l[row] -> row_out[row].
// ---------------------------------------------------------------------------
__global__ __launch_bounds__(BLOCK)
void row_loss_kernel(const float* __restrict__ f, float* __restrict__ row_out) {
  __shared__ float shA[N_STOCKS];   // raw f, then reused: inclusive cumsum of exp(f)
  __shared__ float shB[N_STOCKS];   // inclusive cumsum of exp(-f)
  __shared__ float wtE[8], wtN[8];  // per-wave totals
  __shared__ float part[BLOCK];     // reduction scratch

  const int t   = threadIdx.x;
  const int row = blockIdx.x;
  const float* rowp = f + (size_t)row * N_STOCKS;

#if defined(HAVE_TDM)
  if (t < 32) {  // one TDM descriptor issued by wave 0 (EXEC ignored by TDM)
    const uint64_t gaddr = (uint64_t)(uintptr_t)rowp;
    const uint32_t laddr = (uint32_t)(uintptr_t)(void*)shA;  // low 32b = LDS byte addr
    u32x4 g0;
    g0[0] = 1u;                                    // count=1, user descriptor
    g0[1] = laddr;                                 // lds_addr [63:32]
    g0[2] = (uint32_t)(gaddr & 0xFFFFFFFFu);       // global_addr [95:64]
    g0[3] = (uint32_t)((gaddr >> 32) & 0x01FFFFFFu) | (2u << 30);  // addr[56:32] | type=2
    i32x8 g1;
    g1[0] = (int)(2u << 16);                            // data_size = 4B
    g1[1] = (int)(((unsigned)N_STOCKS & 0xFFFFu) << 16);// tensor_dim0 lo16 @ [63:48]
    g1[2] = (int)(((unsigned)N_STOCKS >> 16) | (1u << 16)); // dim0 hi16 | tensor_dim1=1 lo16
    g1[3] = (int)((unsigned)N_STOCKS << 16);            // tensor_dim1 hi16 | tile_dim0=4096
    g1[4] = 1;                                          // tile_dim1=1, tile_dim2=0
    g1[5] = (int)N_STOCKS;                              // tensor_dim0_stride lo32
    g1[6] = 0;                                          // stride hi16 | dim1_stride lo16
    g1[7] = 0;
    i32x4 z4 = {0, 0, 0, 0};
    i32x8 z8 = {0, 0, 0, 0, 0, 0, 0, 0};
    __builtin_amdgcn_tensor_load_to_lds(g0, g1, z4, z4, z8, 0);
    __builtin_amdgcn_s_wait_tensorcnt(0);
  }
  __syncthreads();
#else
  {
    float4* dstA = (float4*)shA;
    const float4* src = (const float4*)rowp;
#pragma unroll
    for (int v = t; v < N_STOCKS / 4; v += BLOCK) dstA[v] = src[v];
    __syncthreads();
  }
#endif

  // Phase B: per-thread serial scan of 16 contiguous elements (registers)
  const int j0 = t * PER_THR;
  float le[PER_THR], ln[PER_THR];
  float se = 0.f, sn = 0.f, base = 0.f;
#pragma unroll
  for (int k = 0; k < PER_THR; ++k) {
    const float fv = shA[j0 + k];
    base += (j0 + k >= HALF_N) ? fv : -fv;
    const float e = __expf(fv);
    se += e; le[k] = se;
    const float q = __expf(-fv);
    sn += q; ln[k] = sn;
  }

  // wave-level inclusive scans of thread sums, then cross-wave offsets
  const float iE = wave_incl_scan(se);
  const float iN = wave_incl_scan(sn);
  const int lane = t & 31, wid = t >> 5;
  if (lane == 31) { wtE[wid] = iE; wtN[wid] = iN; }
  __syncthreads();
  float offE = iE - se, offN = iN - sn;   // exclusive within wave
  for (int w = 0; w < wid; ++w) { offE += wtE[w]; offN += wtN[w]; }

  // publish global inclusive prefix sums (shA safely reused: region private to t)
#pragma unroll
  for (int k = 0; k < PER_THR; ++k) {
    shA[j0 + k] = le[k] + offE;
    shB[j0 + k] = ln[k] + offN;
  }
  __syncthreads();

  // Phase C: window terms. win(i) = incl[n-1-i] - (i ? incl[i-1] : 0)
  float acc = base;
#pragma unroll
  for (int i = t; i < HALF_N; i += BLOCK) {
    const float we = shA[N_STOCKS - 1 - i] - (i ? shA[i - 1] : 0.f);
    const float wn = shB[N_STOCKS - 1 - i] - (i ? shB[i - 1] : 0.f);
    acc += __logf(we * wn - (float)(N_STOCKS - 2 * i));
  }

  const float total = block_reduce_256(acc, part);
  if (t == 0) row_out[row] = total;
}

// ---------------------------------------------------------------------------
// Final mean over rows (single block, deterministic).
// ---------------------------------------------------------------------------
__global__ __launch_bounds__(BLOCK)
void mean_kernel(const float* __restrict__ row_out, float* __restrict__ out, int nrows) {
  __shared__ float part[BLOCK];
  const int t = threadIdx.x;
  float acc = 0.f;
  for (int i = t; i < nrows; i += BLOCK) acc += row_out[i];
  const float total = block_reduce_256(acc, part);
  if (t == 0) out[0] = total / (float)nrows;
}

extern "C" void kernel_launch(void* const* d_in, const int* in_sizes, int n_in,
                              void* d_out, int out_size, void* d_ws, size_t ws_size,
                              hipStream_t stream) {
  (void)n_in; (void)out_size; (void)ws_size;
  const float* f = (const float*)d_in[0];
  const int nrows = in_sizes[0] / N_STOCKS;     // 8192 for the reference shapes
  float* row_l = (float*)d_ws;                  // nrows floats of scratch
  row_loss_kernel<<<nrows, BLOCK, 0, stream>>>(f, row_l);
  mean_kernel<<<1, BLOCK, 0, stream>>>(row_l, (float*)d_out, nrows);
}